// IntermediateAttentionFusion_30511447671366
// MI455X (gfx1250) — compile-verified
//
#include <hip/hip_runtime.h>
#include <math.h>

// Problem constants (from reference setup_inputs)
#define BB 4
#define TT 2048
#define DD 512
#define HH 8
#define LL 2
#define EPSF 1e-5f
#define SCALEF 0.125f        // dh^-0.5, dh=64
#define NCHUNK (TT / 16)     // 128 chunks of 16 tokens

typedef float v2f __attribute__((ext_vector_type(2)));
typedef float v8f __attribute__((ext_vector_type(8)));

// ---------------------------------------------------------------------------
// Kernel 1: summary — per-token LN stats + masked sum of xhat over tokens.
// One wave (32 lanes) per 16-token chunk. Lane j-slice: d = j*32 + lane.
// Writes per-chunk partial sums P[b][chunk][512] (deterministic reduction
// later in prep kernel).
// ---------------------------------------------------------------------------
__global__ void __launch_bounds__(32)
summary_kernel(const float* __restrict__ x, const int* __restrict__ kpm,
               float* __restrict__ P) {
  const int chunk = blockIdx.x;
  const int b = blockIdx.y;
  const int lane = threadIdx.x;
  const float* xb = x + ((size_t)b * TT + (size_t)chunk * 16) * DD;
  const int* mk = kpm + (size_t)b * TT + (size_t)chunk * 16;

  float acc[16];
#pragma unroll
  for (int j = 0; j < 16; ++j) acc[j] = 0.f;

  for (int t = 0; t < 16; ++t) {
    float v[16];
    float s = 0.f, ss = 0.f;
#pragma unroll
    for (int j = 0; j < 16; ++j) {
      v[j] = xb[t * DD + j * 32 + lane];
      s += v[j];
      ss += v[j] * v[j];
    }
    // butterfly reduce across wave32
    for (int m = 16; m >= 1; m >>= 1) {
      s += __shfl_xor(s, m, 32);
      ss += __shfl_xor(ss, m, 32);
    }
    const float mean = s * (1.f / DD);
    const float var = ss * (1.f / DD) - mean * mean;
    const float rstd = rsqrtf(var + EPSF);
    const float w = (mk[t] == 0) ? rstd : 0.f;
#pragma unroll
    for (int j = 0; j < 16; ++j) acc[j] += w * (v[j] - mean);
  }

  float* Pp = P + ((size_t)b * NCHUNK + chunk) * DD;
#pragma unroll
  for (int j = 0; j < 16; ++j) Pp[j * 32 + lane] = acc[j];
}

// ---------------------------------------------------------------------------
// Kernel 2: prep — per batch, fold everything token-independent:
//   sx[d]   = sum of partials (masked xhat sum of KEY stream)
//   s[d]    = lnG[d]*sx[d] + lnB[d]*cnt            (key-stream LN affine)
//   ksum[o] = sum_d s[d] * Wk[o][d]
//   m[d,h]  = (scale/valid) * sum_{o in head h} Wq[o][d] * ksum[o]
//   mgp[b][n][k] = lnG[k]*m[k,n]  (n<8; n=8..15 zero pad)   -> WMMA B operand
//   mb[b][h] = sum_d lnB[d]*m[d,h]
//   allf[b]  = (cnt==0)
// ---------------------------------------------------------------------------
__global__ void __launch_bounds__(256)
prep_kernel(const float* __restrict__ P, const int* __restrict__ kpm_key,
            const float* __restrict__ lnG, const float* __restrict__ lnB,
            const float* __restrict__ Wk, const float* __restrict__ Wq,
            float* __restrict__ mgp, float* __restrict__ mb,
            int* __restrict__ allf) {
  __shared__ float sx[DD];
  __shared__ float sv[DD];
  __shared__ float ksum[DD];
  __shared__ int cntbuf[256];
  __shared__ float mbpart[256][HH];

  const int b = blockIdx.x;
  const int tid = threadIdx.x;

  // reduce partials over chunks (fixed order -> deterministic)
  for (int d = tid; d < DD; d += 256) {
    float a = 0.f;
    const float* Pp = P + (size_t)b * NCHUNK * DD + d;
    for (int c = 0; c < NCHUNK; ++c) a += Pp[(size_t)c * DD];
    sx[d] = a;
  }

  // count valid key tokens
  int c = 0;
  for (int t = tid; t < TT; t += 256) c += (kpm_key[(size_t)b * TT + t] == 0);
  cntbuf[tid] = c;
  __syncthreads();
  for (int s = 128; s > 0; s >>= 1) {
    if (tid < s) cntbuf[tid] += cntbuf[tid + s];
    __syncthreads();
  }
  const int cnt = cntbuf[0];
  const float fvalid = (cnt > 0) ? (float)cnt : 1.f;
  const float inv = SCALEF / fvalid;
  if (tid == 0) allf[b] = (cnt == 0) ? 1 : 0;

  for (int d = tid; d < DD; d += 256) sv[d] = lnG[d] * sx[d] + lnB[d] * (float)cnt;
  __syncthreads();

  // ksum[o] = s . Wk_row(o)
  for (int o = tid; o < DD; o += 256) {
    const float* wr = Wk + (size_t)o * DD;
    float a = 0.f;
    for (int d = 0; d < DD; ++d) a += sv[d] * wr[d];
    ksum[o] = a;
  }
  __syncthreads();

  // m[d,h], mgp (B-operand layout [n][k]), mb partials
  float mbacc[HH];
#pragma unroll
  for (int h = 0; h < HH; ++h) mbacc[h] = 0.f;

  for (int dd = 0; dd < 2; ++dd) {
    const int d = tid * 2 + dd;
    for (int h = 0; h < HH; ++h) {
      const int ob = h * 64;
      float a = 0.f;
      for (int j = 0; j < 64; ++j) a += Wq[(size_t)(ob + j) * DD + d] * ksum[ob + j];
      const float m = inv * a;
      mgp[((size_t)b * 16 + h) * DD + d] = lnG[d] * m;
      mbacc[h] += lnB[d] * m;
    }
  }
  // zero pad rows n = 8..15 of B operand
  for (int idx = tid; idx < 8 * DD; idx += 256)
    mgp[(size_t)b * 16 * DD + 8 * DD + idx] = 0.f;

#pragma unroll
  for (int h = 0; h < HH; ++h) mbpart[tid][h] = mbacc[h];
  __syncthreads();
  for (int s = 128; s > 0; s >>= 1) {
    if (tid < s) {
#pragma unroll
      for (int h = 0; h < HH; ++h) mbpart[tid][h] += mbpart[tid + s][h];
    }
    __syncthreads();
  }
  if (tid < HH) mb[b * HH + tid] = mbpart[0][tid];
}

// ---------------------------------------------------------------------------
// Kernel 3: update — one wave per 16-token tile.
//   pass1: LN stats + xhat -> LDS
//   WMMA : rel[16 tok x 16] = xhat(16x512) @ mgp(512x16) via
//          128 x v_wmma_f32_16x16x4_f32 (exact fp32)
//   MLP  : gate = sigmoid(W2.relu(W1.rel+b1)+b2), forced 0 if key all-masked
//   pass3: x_out = (1+gate)*x ; emit next-stage masked-xhat partials using
//          exact analytic stats of the scaled row (sum->c*sum, var->c^2*var)
// ---------------------------------------------------------------------------
__global__ void __launch_bounds__(32)
update_kernel(const float* __restrict__ xin, float* __restrict__ xout,
              const int* __restrict__ kpm_self,
              const float* __restrict__ mgp, const float* __restrict__ mb,
              const int* __restrict__ allf,
              const float* __restrict__ W1, const float* __restrict__ b1,
              const float* __restrict__ W2, const float* __restrict__ b2,
              float* __restrict__ P, int emitP) {
  __shared__ float xh[16][516];   // stride 516 avoids bank conflicts on column reads
  __shared__ float relb[16][16];
  __shared__ float meanb[16], sdb[16], varb[16], gateb[16];

  const int chunk = blockIdx.x;
  const int b = blockIdx.y;
  const int lane = threadIdx.x;
  const float* xb = xin + ((size_t)b * TT + (size_t)chunk * 16) * DD;
  float* ob = xout + ((size_t)b * TT + (size_t)chunk * 16) * DD;

  // ---- pass 1: LN stats + xhat into LDS ----
  for (int t = 0; t < 16; ++t) {
    float v[16];
    float s = 0.f, ss = 0.f;
#pragma unroll
    for (int j = 0; j < 16; ++j) {
      v[j] = xb[t * DD + j * 32 + lane];
      s += v[j];
      ss += v[j] * v[j];
    }
    for (int m = 16; m >= 1; m >>= 1) {
      s += __shfl_xor(s, m, 32);
      ss += __shfl_xor(ss, m, 32);
    }
    const float mean = s * (1.f / DD);
    const float var = ss * (1.f / DD) - mean * mean;
    const float sd = sqrtf(var + EPSF);
    const float rstd = 1.f / sd;
#pragma unroll
    for (int j = 0; j < 16; ++j) xh[t][j * 32 + lane] = (v[j] - mean) * rstd;
    if (lane == 0) { meanb[t] = mean; sdb[t] = sd; varb[t] = var; }
  }
  __syncthreads();

  // ---- WMMA: rel = xhat @ mgp  (f32 16x16x4, K=512 -> 128 steps) ----
  const bool hi = lane >= 16;
  const int row = lane & 15;
  const float* mg = mgp + (size_t)b * 16 * DD;  // layout [n][k]
  v8f acc = {};
  for (int kk = 0; kk < 128; ++kk) {
    const int k0 = kk * 4 + (hi ? 2 : 0);
    v2f a, bmat;
    a.x = xh[row][k0];
    a.y = xh[row][k0 + 1];
    bmat.x = mg[(size_t)row * DD + k0];
    bmat.y = mg[(size_t)row * DD + k0 + 1];
    acc = __builtin_amdgcn_wmma_f32_16x16x4_f32(false, a, false, bmat,
                                                (short)0, acc, false, false);
  }
  // scatter C tile (VGPR r: lanes0-15 -> M=r, lanes16-31 -> M=r+8; N=row)
#pragma unroll
  for (int r = 0; r < 8; ++r) relb[(hi ? 8 : 0) + r][row] = acc[r];
  __syncthreads();

  // ---- gate MLP: lanes 0..15 each handle one token ----
  if (lane < 16) {
    const int t = lane;
    float r8[HH];
#pragma unroll
    for (int h = 0; h < HH; ++h) r8[h] = relb[t][h] + mb[b * HH + h];
    float z = b2[0];
    for (int e = 0; e < 16; ++e) {
      float h1 = b1[e];
#pragma unroll
      for (int h = 0; h < HH; ++h) h1 += r8[h] * W1[e * HH + h];
      h1 = fmaxf(h1, 0.f);
      z += W2[e] * h1;
    }
    const float g = allf[b] ? 0.f : 1.f / (1.f + expf(-z));
    gateb[t] = g;
  }
  __syncthreads();

  // ---- pass 3: scale rows, emit next-stage masked-xhat partials ----
  float accj[16];
#pragma unroll
  for (int j = 0; j < 16; ++j) accj[j] = 0.f;
  const int* mk = kpm_self + (size_t)b * TT + (size_t)chunk * 16;

  for (int t = 0; t < 16; ++t) {
    const float c = 1.f + gateb[t];
    const float mean = meanb[t];
    const float sd = sdb[t];
    const float var = varb[t];
    // x_new = c*x  ->  mean'=c*mean, var'=c^2*var (exact), rstd' with eps
    const float rstd2 = rsqrtf(c * c * var + EPSF);
    const float coeff = c * sd * rstd2;  // xhat_new = coeff * xhat_old
    const float w = (mk[t] == 0) ? coeff : 0.f;
#pragma unroll
    for (int j = 0; j < 16; ++j) {
      const float xhv = xh[t][j * 32 + lane];
      const float xv = xhv * sd + mean;
      ob[t * DD + j * 32 + lane] = c * xv;
      accj[j] += w * xhv;
    }
  }

  if (emitP) {
    float* Pp = P + ((size_t)b * NCHUNK + chunk) * DD;
#pragma unroll
    for (int j = 0; j < 16; ++j) Pp[j * 32 + lane] = accj[j];
  }
}

// ---------------------------------------------------------------------------
// Host launcher
// ---------------------------------------------------------------------------
extern "C" void kernel_launch(void* const* d_in, const int* in_sizes, int n_in,
                              void* d_out, int out_size, void* d_ws, size_t ws_size,
                              hipStream_t stream) {
  (void)in_sizes; (void)n_in; (void)out_size; (void)ws_size;

  const float* x_vid = (const float*)d_in[0];
  const float* x_aud = (const float*)d_in[1];
  const int* vid_kpm = (const int*)d_in[2];
  const int* aud_kpm = (const int*)d_in[3];
  const float* Wq_vid = (const float*)d_in[4];
  const float* Wk_a4v = (const float*)d_in[5];
  const float* Wq_aud = (const float*)d_in[6];
  const float* Wk_v4a = (const float*)d_in[7];
  const float* vgW1 = (const float*)d_in[8];
  const float* vgb1 = (const float*)d_in[9];
  const float* vgW2 = (const float*)d_in[10];
  const float* vgb2 = (const float*)d_in[11];
  const float* agW1 = (const float*)d_in[12];
  const float* agb1 = (const float*)d_in[13];
  const float* agW2 = (const float*)d_in[14];
  const float* agb2 = (const float*)d_in[15];
  const float* vlnG = (const float*)d_in[16];
  const float* vlnB = (const float*)d_in[17];
  const float* alnG = (const float*)d_in[18];
  const float* alnB = (const float*)d_in[19];

  float* OV = (float*)d_out;
  float* OA = OV + (size_t)BB * TT * DD;

  // workspace layout
  float* P = (float*)d_ws;                               // B*NCHUNK*D floats
  float* mgp = P + (size_t)BB * NCHUNK * DD;             // B*16*D floats
  float* mb = mgp + (size_t)BB * 16 * DD;                // B*8 floats
  int* allf = (int*)(mb + (size_t)BB * HH);              // B ints

  const dim3 gs(NCHUNK, BB);
  const size_t wOff = (size_t)DD * DD;   // per-layer D*D weight offset
  const size_t lnOff = DD;

  // S0: summary of aud stream (key side of layer-0 dir-1)
  summary_kernel<<<gs, 32, 0, stream>>>(x_aud, aud_kpm, P);

  // L0 dir1: vid gated by aud (vid LN params applied to BOTH streams)
  prep_kernel<<<BB, 256, 0, stream>>>(P, aud_kpm, vlnG, vlnB, Wk_a4v, Wq_vid,
                                      mgp, mb, allf);
  update_kernel<<<gs, 32, 0, stream>>>(x_vid, OV, vid_kpm, mgp, mb, allf,
                                       vgW1, vgb1, vgW2, vgb2, P, 1);

  // L0 dir2: aud gated by updated vid (aud LN params for both)
  prep_kernel<<<BB, 256, 0, stream>>>(P, vid_kpm, alnG, alnB, Wk_v4a, Wq_aud,
                                      mgp, mb, allf);
  update_kernel<<<gs, 32, 0, stream>>>(x_aud, OA, aud_kpm, mgp, mb, allf,
                                       agW1, agb1, agW2, agb2, P, 1);

  // L1 dir1
  prep_kernel<<<BB, 256, 0, stream>>>(P, aud_kpm, vlnG + lnOff, vlnB + lnOff,
                                      Wk_a4v + wOff, Wq_vid + wOff, mgp, mb, allf);
  update_kernel<<<gs, 32, 0, stream>>>(OV, OV, vid_kpm, mgp, mb, allf,
                                       vgW1 + 16 * HH, vgb1 + 16, vgW2 + 16,
                                       vgb2 + 1, P, 1);

  // L1 dir2 (no further summary needed)
  prep_kernel<<<BB, 256, 0, stream>>>(P, vid_kpm, alnG + lnOff, alnB + lnOff,
                                      Wk_v4a + wOff, Wq_aud + wOff, mgp, mb, allf);
  update_kernel<<<gs, 32, 0, stream>>>(OA, OA, aud_kpm, mgp, mb, allf,
                                       agW1 + 16 * HH, agb1 + 16, agW2 + 16,
                                       agb2 + 1, P, 0);
}